// CrossModalRouter_35957466202213
// MI455X (gfx1250) — compile-verified
//
#include <hip/hip_runtime.h>
#include <hip/hip_bf16.h>

typedef float v2f __attribute__((ext_vector_type(2)));
typedef float v4f __attribute__((ext_vector_type(4)));
typedef float v8f __attribute__((ext_vector_type(8)));

#define IN_CH   1024
#define MID     64
#define NE      8
#define BATCH   64
#define HW      4096   // 64*64 spatial

// ---------------------------------------------------------------------------
// Kernel 1: global average pool  [B,C,H,W] -> [B*C]
// One wave32 per (b,c) plane of 4096 floats. 128-bit streaming (non-temporal)
// loads, fully coalesced; wave shuffle reduction. 65536 planes / 8 waves per
// block = 8192 blocks. Pure HBM streaming, ~1 GiB read (~46 us at 23.3 TB/s).
// ---------------------------------------------------------------------------
__global__ __launch_bounds__(256) void cmr_gap_kernel(
    const float* __restrict__ x, float* __restrict__ gf)
{
    const int wave  = threadIdx.x >> 5;
    const int lane  = threadIdx.x & 31;
    const int plane = blockIdx.x * 8 + wave;          // < 65536

    const v4f* p = (const v4f*)(x + (size_t)plane * HW);
    float s = 0.0f;
#pragma unroll
    for (int it = 0; it < 32; ++it) {
        v4f v = __builtin_nontemporal_load(&p[it * 32 + lane]);
        s += (v.x + v.y) + (v.z + v.w);
    }
#pragma unroll
    for (int off = 16; off > 0; off >>= 1)
        s += __shfl_xor(s, off, 32);
    if (lane == 0) gf[plane] = s * (1.0f / (float)HW);
}

// ---------------------------------------------------------------------------
// Kernel 2: router MLP + noisy top-2 routing. Single block, 4 waves.
// hidden = silu(gf @ W1 + b1) via V_WMMA_F32_16X16X4_F32:
//   each wave owns rows [wave*16, wave*16+16) and all four 16-col tiles,
//   accumulating over K=1024 in steps of 4.
// ---------------------------------------------------------------------------
__global__ __launch_bounds__(128) void cmr_router_kernel(
    const float* __restrict__ gf,   const float* __restrict__ noise,
    const float* __restrict__ W1,   const float* __restrict__ b1,
    const float* __restrict__ W2,   const float* __restrict__ b2,
    float* __restrict__ out)
{
    __shared__ float hidden[BATCH][MID];
    __shared__ int   cnt[NE];

    const int lane = threadIdx.x & 31;
    const int wave = threadIdx.x >> 5;
    const int m0   = wave * 16;          // row-block of this wave
    const int half = lane >> 4;          // 0: lanes 0-15, 1: lanes 16-31
    const int l16  = lane & 15;

    v8f acc0 = {}, acc1 = {}, acc2 = {}, acc3 = {};

    // A fragment (16x4 f32, ISA 7.12.2): lane<16 holds K=k..k+1 of row M=l16,
    // lane>=16 holds K=k+2..k+3 of row M=l16.
    const float* arow = gf + (m0 + l16) * IN_CH + half * 2;

    for (int k = 0; k < IN_CH; k += 4) {
        v2f a;
        a.x = arow[k + 0];
        a.y = arow[k + 1];

        // B fragment (4x16 f32): VGPR0 = row K=k+2*half (N=l16),
        //                        VGPR1 = row K=k+2*half+1.
        const float* brow = W1 + (size_t)(k + 2 * half) * MID + l16;
        v2f bb0, bb1, bb2, bb3;
        bb0.x = brow[0];       bb0.y = brow[MID + 0];
        bb1.x = brow[16];      bb1.y = brow[MID + 16];
        bb2.x = brow[32];      bb2.y = brow[MID + 32];
        bb3.x = brow[48];      bb3.y = brow[MID + 48];

        acc0 = __builtin_amdgcn_wmma_f32_16x16x4_f32(false, a, false, bb0,
                                                     (short)0, acc0, false, false);
        acc1 = __builtin_amdgcn_wmma_f32_16x16x4_f32(false, a, false, bb1,
                                                     (short)0, acc1, false, false);
        acc2 = __builtin_amdgcn_wmma_f32_16x16x4_f32(false, a, false, bb2,
                                                     (short)0, acc2, false, false);
        acc3 = __builtin_amdgcn_wmma_f32_16x16x4_f32(false, a, false, bb3,
                                                     (short)0, acc3, false, false);
    }

    // D layout: VGPR r holds M = m0 + r + 8*half, N = n0 + l16.
#pragma unroll
    for (int r = 0; r < 8; ++r) {
        const int row = m0 + r + 8 * half;
        float v;
        v = acc0[r] + b1[ 0 + l16]; hidden[row][ 0 + l16] = v / (1.0f + __expf(-v));
        v = acc1[r] + b1[16 + l16]; hidden[row][16 + l16] = v / (1.0f + __expf(-v));
        v = acc2[r] + b1[32 + l16]; hidden[row][32 + l16] = v / (1.0f + __expf(-v));
        v = acc3[r] + b1[48 + l16]; hidden[row][48 + l16] = v / (1.0f + __expf(-v));
    }

    if (threadIdx.x < NE) cnt[threadIdx.x] = 0;
    __syncthreads();

    // Second layer [64,64]@[64,8] + routing: one thread per batch row.
    if (threadIdx.x < BATCH) {
        const int b = threadIdx.x;
        float logit[NE];
#pragma unroll
        for (int e = 0; e < NE; ++e) {
            float s = b2[e];
            for (int m = 0; m < MID; ++m)
                s += hidden[b][m] * W2[m * NE + e];
            logit[e] = s;
        }
        float nz[NE];
#pragma unroll
        for (int e = 0; e < NE; ++e)
            nz[e] = logit[e] + noise[b * NE + e] * 2.0f;

        // top-2 on noisy logits; strict '>' keeps JAX's lowest-index tie-break
        int i0 = 0;
#pragma unroll
        for (int e = 1; e < NE; ++e) if (nz[e] > nz[i0]) i0 = e;
        int i1 = (i0 == 0) ? 1 : 0;
#pragma unroll
        for (int e = 0; e < NE; ++e)
            if (e != i0 && nz[e] > nz[i1]) i1 = e;

        // softmax over the two *clean* logits
        const float l0 = logit[i0], l1 = logit[i1];
        const float mx = fmaxf(l0, l1);
        const float e0 = __expf(l0 - mx), e1 = __expf(l1 - mx);
        const float inv = 1.0f / (e0 + e1);

        out[b * 2 + 0]       = e0 * inv;
        out[b * 2 + 1]       = e1 * inv;
        out[128 + b * 2 + 0] = (float)i0;
        out[128 + b * 2 + 1] = (float)i1;
        atomicAdd(&cnt[i0], 1);
        atomicAdd(&cnt[i1], 1);
    }
    __syncthreads();
    if (threadIdx.x < NE)
        out[256 + threadIdx.x] = (float)cnt[threadIdx.x];
}

extern "C" void kernel_launch(void* const* d_in, const int* in_sizes, int n_in,
                              void* d_out, int out_size, void* d_ws, size_t ws_size,
                              hipStream_t stream)
{
    (void)in_sizes; (void)n_in; (void)out_size; (void)ws_size;
    const float* x     = (const float*)d_in[0];
    const float* noise = (const float*)d_in[1];
    const float* W1    = (const float*)d_in[2];
    const float* b1    = (const float*)d_in[3];
    const float* W2    = (const float*)d_in[4];
    const float* b2    = (const float*)d_in[5];

    float* gf = (float*)d_ws;   // [64*1024] pooled features

    cmr_gap_kernel<<<(BATCH * IN_CH) / 8, 256, 0, stream>>>(x, gf);
    cmr_router_kernel<<<1, 128, 0, stream>>>(gf, noise, W1, b1, W2, b2,
                                             (float*)d_out);
}